// DiouLoss_26766236189167
// MI455X (gfx1250) — compile-verified
//
#include <hip/hip_runtime.h>
#include <stdint.h>

#define BATCH 32
#define HH 256
#define WW 256
#define HWSZ (HH * WW)
#define TOPK 500
#define CAP 8192           // >= ~7300 expected local maxima; padded with 0-keys
#define TROWS 66           // 64 compute rows + 2 halo rows

typedef unsigned int uint32;
typedef unsigned long long u64;
typedef __attribute__((ext_vector_type(4))) uint32 v4u;
typedef __attribute__((ext_vector_type(8))) uint32 v8u;
typedef __attribute__((ext_vector_type(2))) float v2f;
typedef __attribute__((ext_vector_type(8))) float v8f;

// ---------------------------------------------------------------------------
// TDM: DMA a block of `nrows` full 256-float rows from global into LDS.
// Builds a 2D Tensor DMA Descriptor (D# groups 0 and 1) per cdna5 ISA ch.8
// and issues TENSOR_LOAD_TO_LDS via inline asm (portable across toolchains).
// ---------------------------------------------------------------------------
__device__ __forceinline__ void tdm_load_rows_to_lds(const float* gsrc,
                                                     uint32 lds_byte_off,
                                                     uint32 nrows) {
  unsigned long long ga = (unsigned long long)(uintptr_t)gsrc;
  v4u g0;
  g0[0] = 1u;                                          // count=1, user D#, no gather
  g0[1] = lds_byte_off;                                // lds_addr (bytes)
  g0[2] = (uint32)(ga & 0xFFFFFFFFull);                // global_addr[31:0]
  g0[3] = (uint32)((ga >> 32) & 0x01FFFFFFu)           // global_addr[56:32]
          | (2u << 30);                                // type = 2 ("image")
  v8u g1;
  g1[0] = (2u << 16);                                  // wg_mask=0, data_size=2 (4B)
  g1[1] = ((uint32)WW & 0xFFFFu) << 16;                // tensor_dim0[15:0]
  g1[2] = ((uint32)WW >> 16) | ((nrows & 0xFFFFu) << 16); // dim0[31:16] | tensor_dim1 lo
  g1[3] = (nrows >> 16) | ((uint32)WW << 16);          // tensor_dim1 hi | tile_dim0=256
  g1[4] = (nrows & 0xFFFFu);                           // tile_dim1 = nrows, tile_dim2=0
  g1[5] = (uint32)WW;                                  // tensor_dim0_stride = 256
  g1[6] = 0u;                                          // stride0 hi | stride1 lo
  g1[7] = 0u;                                          // stride1 hi
  asm volatile("tensor_load_to_lds %0, %1" : : "s"(g0), "s"(g1) : "memory");
}

// ---------------------------------------------------------------------------
// Kernel 1: per-(batch, heatmap) NMS + top-500 with sorted output.
// One 1024-thread (32 wave32) workgroup per block. Image streamed in 4 row
// chunks through LDS via the TDM; strict 3x3 local maxima are compacted into
// packed 64-bit keys and bitonic-sorted descending in LDS.
// key = (float_bits(value) << 32) | ~index  -> value desc, index asc on ties.
// ---------------------------------------------------------------------------
__global__ __launch_bounds__(1024) void nms_topk_kernel(
    const float* __restrict__ pred_hm, const float* __restrict__ gt_hm,
    float* __restrict__ ws) {
  __shared__ float s_tile[TROWS * WW];   // 67.5 KB
  __shared__ u64 s_keys[CAP];            // 64   KB
  __shared__ int s_cnt;

  const int tid = threadIdx.x;
  const int b = blockIdx.x >> 1;
  const int which = blockIdx.x & 1;      // 0 = pred_hm, 1 = gt_hm
  const float* img = (which ? gt_hm : pred_hm) + (size_t)b * HWSZ;

  if (tid == 0) s_cnt = 0;
  __syncthreads();

  for (int chunk = 0; chunk < 4; ++chunk) {
    const int top = chunk * 64 - 1;                 // first halo row (may be -1)
    const int lo = top < 0 ? 0 : top;               // clamped first loaded row
    const int hi = min(chunk * 64 + 64, HH - 1);    // clamped last loaded row
    const int nrows = hi - lo + 1;                  // 65 or 66
    const int lds_row0 = lo - top;                  // 1 for chunk 0, else 0
    __syncthreads();                                // prior chunk fully consumed
    if (tid < 32) {                                 // wave 0 drives the DMA
      tdm_load_rows_to_lds(
          img + (size_t)lo * WW,
          (uint32)(uintptr_t)(&s_tile[lds_row0 * WW]), (uint32)nrows);
      __builtin_amdgcn_s_wait_tensorcnt(0);
    }
    __syncthreads();

    for (int p = tid; p < 64 * WW; p += (int)blockDim.x) {
      const int r = chunk * 64 + (p >> 8);
      const int c = p & (WW - 1);
      const int lr = (p >> 8) + 1;                  // LDS row of pixel r
      const float* rowm = &s_tile[(lr - 1) * WW];
      const float* row0 = &s_tile[lr * WW];
      const float* rowp = &s_tile[(lr + 1) * WW];
      const float v = row0[c];
      float mx = v;
      const bool cl = (c > 0), cr = (c < WW - 1);
      if (r > 0) {
        mx = fmaxf(mx, rowm[c]);
        if (cl) mx = fmaxf(mx, rowm[c - 1]);
        if (cr) mx = fmaxf(mx, rowm[c + 1]);
      }
      if (cl) mx = fmaxf(mx, row0[c - 1]);
      if (cr) mx = fmaxf(mx, row0[c + 1]);
      if (r < HH - 1) {
        mx = fmaxf(mx, rowp[c]);
        if (cl) mx = fmaxf(mx, rowp[c - 1]);
        if (cr) mx = fmaxf(mx, rowp[c + 1]);
      }
      if (v == mx && v > 0.0f) {                    // strict local max survives NMS
        const int pos = atomicAdd(&s_cnt, 1);
        if (pos < CAP) {
          const uint32 idx = (uint32)(r * WW + c);
          s_keys[pos] = ((u64)__float_as_uint(v) << 32) | (u64)(~idx);
        }
      }
    }
  }
  __syncthreads();

  int cnt = s_cnt;
  if (cnt > CAP) cnt = CAP;
  for (int i = cnt + tid; i < CAP; i += (int)blockDim.x) s_keys[i] = 0ull;

  // Bitonic sort, descending.
  for (uint32 kk = 2; kk <= CAP; kk <<= 1) {
    for (uint32 j = kk >> 1; j > 0; j >>= 1) {
      __syncthreads();
      for (uint32 i = tid; i < CAP; i += blockDim.x) {
        const uint32 ixj = i ^ j;
        if (ixj > i) {
          const u64 a = s_keys[i], c2 = s_keys[ixj];
          const bool up = ((i & kk) == 0);
          if (up ? (a < c2) : (a > c2)) { s_keys[i] = c2; s_keys[ixj] = a; }
        }
      }
    }
  }
  __syncthreads();

  // ws layout: [xs_pred][ys_pred][xs_gt][ys_gt], each BATCH*TOPK floats.
  float* xs_out = ws + (size_t)(which * 2 + 0) * (BATCH * TOPK) + b * TOPK;
  float* ys_out = ws + (size_t)(which * 2 + 1) * (BATCH * TOPK) + b * TOPK;
  if (tid < TOPK) {
    const u64 key = s_keys[tid];
    const uint32 idx = ~((uint32)(key & 0xFFFFFFFFull));
    xs_out[tid] = (float)(idx & (WW - 1));
    ys_out[tid] = (float)(idx >> 8);
  }
}

// ---------------------------------------------------------------------------
// Wave32 sum reduction via V_WMMA_F32_16X16X4_F32:
// A (16x4, 2 VGPRs/lane) holds {partial, 0}; B = ones(4x16). D[m][n] =
// partial[m] + partial[m+16]; summing a lane's 8 D VGPRs gives a half-wave
// sum, completed by one shfl_xor across the lane-16 boundary.
// ---------------------------------------------------------------------------
__device__ __forceinline__ float wave_sum(float v) {
#if __has_builtin(__builtin_amdgcn_wmma_f32_16x16x4_f32)
  v2f a; a[0] = v; a[1] = 0.0f;
  v2f ones; ones[0] = 1.0f; ones[1] = 1.0f;
  v8f c = {0.f, 0.f, 0.f, 0.f, 0.f, 0.f, 0.f, 0.f};
  c = __builtin_amdgcn_wmma_f32_16x16x4_f32(false, a, false, ones,
                                            (short)0, c, false, false);
  float s = c[0] + c[1] + c[2] + c[3] + c[4] + c[5] + c[6] + c[7];
  s += __shfl_xor(s, 16, 32);
  return s;
#else
  for (int off = 16; off > 0; off >>= 1) v += __shfl_xor(v, off, 32);
  return v;
#endif
}

__device__ __forceinline__ void assemble(float xs, float ys, const float* wh,
                                         float m, float* o) {
  const float om = 1.0f - m;
  o[0] = xs;
  o[1] = ys;
  o[2] = (xs + wh[0]) * m + xs * om;
  o[3] = (ys + wh[1]) * m + (ys - wh[9] * 0.5f) * om;
  o[4] = (xs + wh[2]) * m + (xs + wh[8] * 0.5f) * om;
  o[5] = (ys + wh[3]) * m + ys * om;
  o[6] = (xs + wh[4]) * m + xs * om;
  o[7] = (ys + wh[5]) * m + (ys + wh[9] * 0.5f) * om;
  o[8] = (xs + wh[6]) * m + (xs - wh[8] * 0.5f) * om;
  o[9] = (ys + wh[7]) * m + ys * om;
}

// ---------------------------------------------------------------------------
// Kernel 2: one block per batch; thread k gathers at gt_ind[b,k], assembles
// pred/gt boxes, smooth-L1, masked; block sum written deterministically.
// ---------------------------------------------------------------------------
__global__ __launch_bounds__(512) void loss_partial_kernel(
    const float* __restrict__ pred_wh, const float* __restrict__ pred_reg,
    const float* __restrict__ pred_cls_theta, const float* __restrict__ gt_wh,
    const float* __restrict__ gt_reg, const float* __restrict__ gt_cls_theta,
    const int* __restrict__ gt_ind, const int* __restrict__ gt_reg_mask,
    const float* __restrict__ ws_xy, float* __restrict__ bsums) {
  const int b = blockIdx.x;
  const int k = threadIdx.x;
  float sumE = 0.0f, sumM = 0.0f;
  if (k < TOPK) {
    const int bk = b * TOPK + k;
    const int ind = gt_ind[bk];
    const float* xs0p = ws_xy;
    const float* ys0p = ws_xy + BATCH * TOPK;
    const float* xs0g = ws_xy + 2 * BATCH * TOPK;
    const float* ys0g = ws_xy + 3 * BATCH * TOPK;

    const float xs = xs0p[bk] + pred_reg[((size_t)b * 2 + 0) * HWSZ + ind];
    const float ys = ys0p[bk] + pred_reg[((size_t)b * 2 + 1) * HWSZ + ind];
    const float gxs = xs0g[bk] + gt_reg[(size_t)bk * 2 + 0];
    const float gys = ys0g[bk] + gt_reg[(size_t)bk * 2 + 1];

    float wh[10], gwh[10];
#pragma unroll
    for (int c = 0; c < 10; ++c) {
      wh[c] = pred_wh[((size_t)b * 10 + c) * HWSZ + ind];
      gwh[c] = gt_wh[(size_t)bk * 10 + c];
    }
    const float m = (pred_cls_theta[(size_t)b * HWSZ + ind] > 0.8f) ? 1.0f : 0.0f;
    const float gm = (gt_cls_theta[bk] > 0.8f) ? 1.0f : 0.0f;

    float pp[10], tt[10];
    assemble(xs, ys, wh, m, pp);
    assemble(gxs, gys, gwh, gm, tt);

    const float rm = (float)gt_reg_mask[bk];
    float e = 0.0f;
#pragma unroll
    for (int i = 0; i < 10; ++i) {
      const float d = fabsf(pp[i] - tt[i]);
      e += (d < 1.0f) ? (0.5f * d * d) : (d - 0.5f);
    }
    sumE = rm * e;
    sumM = rm * 10.0f;
  }
  // EXEC is reconverged (all lanes active) before the WMMA reductions.
  const float wE = wave_sum(sumE);
  const float wM = wave_sum(sumM);
  __shared__ float shE[16], shM[16];
  const int wave = threadIdx.x >> 5;
  const int lane = threadIdx.x & 31;
  if (lane == 0) { shE[wave] = wE; shM[wave] = wM; }
  __syncthreads();
  if (threadIdx.x == 0) {
    float tE = 0.0f, tM = 0.0f;
    for (int w = 0; w < (int)(blockDim.x >> 5); ++w) { tE += shE[w]; tM += shM[w]; }
    bsums[2 * b + 0] = tE;
    bsums[2 * b + 1] = tM;
  }
}

__global__ void finalize_kernel(const float* __restrict__ bsums,
                                float* __restrict__ out) {
  if (threadIdx.x == 0 && blockIdx.x == 0) {
    float tE = 0.0f, tM = 0.0f;
    for (int b = 0; b < BATCH; ++b) { tE += bsums[2 * b]; tM += bsums[2 * b + 1]; }
    out[0] = (tM > 0.0f) ? (tE / fmaxf(tM, 1.0f)) : 0.0f;
  }
}

extern "C" void kernel_launch(void* const* d_in, const int* in_sizes, int n_in,
                              void* d_out, int out_size, void* d_ws,
                              size_t ws_size, hipStream_t stream) {
  const float* pred_hm        = (const float*)d_in[0];
  const float* pred_wh        = (const float*)d_in[1];
  const float* pred_reg       = (const float*)d_in[2];
  const float* pred_cls_theta = (const float*)d_in[3];
  const float* gt_hm          = (const float*)d_in[4];
  const float* gt_wh          = (const float*)d_in[5];
  const float* gt_reg         = (const float*)d_in[6];
  const float* gt_cls_theta   = (const float*)d_in[7];
  const int*   gt_ind         = (const int*)d_in[8];
  const int*   gt_reg_mask    = (const int*)d_in[9];

  float* ws = (float*)d_ws;                      // 4*B*K floats of xs/ys
  float* bsums = ws + 4 * BATCH * TOPK;          // + 2*B floats of block sums

  nms_topk_kernel<<<BATCH * 2, 1024, 0, stream>>>(pred_hm, gt_hm, ws);
  loss_partial_kernel<<<BATCH, 512, 0, stream>>>(
      pred_wh, pred_reg, pred_cls_theta, gt_wh, gt_reg, gt_cls_theta, gt_ind,
      gt_reg_mask, ws, bsums);
  finalize_kernel<<<1, 32, 0, stream>>>(bsums, (float*)d_out);
}